// GCNConv__19335942766940
// MI455X (gfx1250) — compile-verified
//
#include <hip/hip_runtime.h>
#include <hip/hip_bf16.h>

#define N_NODES 50000
#define N_EDGES 800000
#define DIM 128
#define LDS_STRIDE 132  // padded row stride (floats): bank step 4 -> conflict-free

typedef float v2f __attribute__((ext_vector_type(2)));
typedef float v8f __attribute__((ext_vector_type(8)));

// ---------------------------------------------------------------------------
// 1) zero output accumulator + degree counters
// ---------------------------------------------------------------------------
__global__ __launch_bounds__(256) void gcn_init(float* __restrict__ out,
                                                unsigned int* __restrict__ deg) {
    int idx = blockIdx.x * blockDim.x + threadIdx.x;
    if (idx < N_NODES * DIM) out[idx] = 0.0f;
    if (idx < N_NODES) deg[idx] = 0u;
}

// ---------------------------------------------------------------------------
// 2) integer degree count over edge targets (exact)
// ---------------------------------------------------------------------------
__global__ __launch_bounds__(256) void gcn_degree(const long long* __restrict__ col,
                                                  unsigned int* __restrict__ deg) {
    int e = blockIdx.x * blockDim.x + threadIdx.x;
    if (e >= N_EDGES) return;
    atomicAdd(&deg[(int)col[e]], 1u);
}

// ---------------------------------------------------------------------------
// 3) dinv = rsqrt(deg + 1)   (+1 = self loop)
// ---------------------------------------------------------------------------
__global__ __launch_bounds__(256) void gcn_dinv(const unsigned int* __restrict__ deg,
                                                float* __restrict__ dinv) {
    int i = blockIdx.x * blockDim.x + threadIdx.x;
    if (i >= N_NODES) return;
    dinv[i] = __frsqrt_rn((float)(deg[i] + 1u));
}

// ---------------------------------------------------------------------------
// 4) xw = x @ W via V_WMMA_F32_16X16X4_F32 (fp32 in/out, wave32)
//    Block = 256 threads = 8 waves. Block owns one 16-row M tile;
//    wave w computes the 16-col N tile w. A tile staged in LDS.
// ---------------------------------------------------------------------------
__global__ __launch_bounds__(256) void gcn_gemm_wmma(const float* __restrict__ x,
                                                     const float* __restrict__ W,
                                                     float* __restrict__ xw) {
    __shared__ float As[16 * LDS_STRIDE];

    const int row0 = blockIdx.x * 16;
    const int tid  = threadIdx.x;

    // cooperative load of 16x128 A tile (coalesced), padded LDS stride
    for (int i = tid; i < 16 * DIM; i += 256) {
        int r = i >> 7;        // /128
        int c = i & (DIM - 1);
        As[r * LDS_STRIDE + c] = x[(size_t)(row0 + r) * DIM + c];
    }
    __syncthreads();

    const int wave  = tid >> 5;          // 0..7 -> N tile
    const int lane  = tid & 31;
    const int n0    = wave * 16;
    const int lhalf = lane & 15;         // M index for A, N index for B
    const int khi   = (lane >> 4) << 1;  // 0 for lanes 0-15, 2 for lanes 16-31

    v8f c = {0.f, 0.f, 0.f, 0.f, 0.f, 0.f, 0.f, 0.f};

    for (int k = 0; k < DIM; k += 4) {
        v2f a, b;
        // A 16x4: lane<16 holds K=k..k+1 of row lhalf; lane>=16 holds K=k+2..k+3
        a.x = As[lhalf * LDS_STRIDE + k + khi];
        a.y = As[lhalf * LDS_STRIDE + k + khi + 1];
        // B 4x16: row K striped across lanes (coalesced from L2-resident W)
        b.x = W[(size_t)(k + khi) * DIM + n0 + lhalf];
        b.y = W[(size_t)(k + khi + 1) * DIM + n0 + lhalf];
        c = __builtin_amdgcn_wmma_f32_16x16x4_f32(
                /*neg_a=*/false, a, /*neg_b=*/false, b,
                /*c_mod=*/(short)0, c, /*reuse_a=*/false, /*reuse_b=*/false);
    }

    // C/D layout: VGPR r -> (M=r, lanes 0-15) and (M=r+8, lanes 16-31)
    const int ncol  = n0 + lhalf;
    const int mbase = row0 + ((lane >> 4) << 3);
#pragma unroll
    for (int r = 0; r < 8; ++r) {
        xw[(size_t)(mbase + r) * DIM + ncol] = c[r];
    }
}

// ---------------------------------------------------------------------------
// 5) edge scatter: out[col] += xw[row] * dinv[row]*dinv[col]
//    One edge per 32-lane wave; each lane moves a float4 (full 128-col row).
// ---------------------------------------------------------------------------
__global__ __launch_bounds__(256) void gcn_scatter(const long long* __restrict__ row,
                                                   const long long* __restrict__ col,
                                                   const float* __restrict__ dinv,
                                                   const float* __restrict__ xw,
                                                   float* __restrict__ out) {
    int gid  = blockIdx.x * blockDim.x + threadIdx.x;
    int e    = gid >> 5;
    int lane = gid & 31;
    if (e >= N_EDGES) return;

    int r = (int)row[e];
    int c = (int)col[e];
    float w = dinv[r] * dinv[c];

    const float4 v = ((const float4*)(xw + (size_t)r * DIM))[lane];
    float* dst = out + (size_t)c * DIM + lane * 4;
    unsafeAtomicAdd(dst + 0, v.x * w);
    unsafeAtomicAdd(dst + 1, v.y * w);
    unsafeAtomicAdd(dst + 2, v.z * w);
    unsafeAtomicAdd(dst + 3, v.w * w);
}

// ---------------------------------------------------------------------------
// 6) out = relu(out + xw*dinv^2 + b)   (self-loop folded in, bias, ReLU)
// ---------------------------------------------------------------------------
__global__ __launch_bounds__(256) void gcn_finalize(const float* __restrict__ xw,
                                                    const float* __restrict__ dinv,
                                                    const float* __restrict__ b,
                                                    float* __restrict__ out) {
    int idx = blockIdx.x * blockDim.x + threadIdx.x;
    if (idx >= N_NODES * DIM) return;
    int n = idx >> 7;          // /128
    int d = idx & (DIM - 1);
    float di = dinv[n];
    float v  = out[idx] + xw[idx] * (di * di) + b[d];
    out[idx] = fmaxf(v, 0.0f);
}

// ---------------------------------------------------------------------------
extern "C" void kernel_launch(void* const* d_in, const int* in_sizes, int n_in,
                              void* d_out, int out_size, void* d_ws, size_t ws_size,
                              hipStream_t stream) {
    const float*     x  = (const float*)d_in[0];
    const long long* ei = (const long long*)d_in[1];  // int64 [2, E] row-major
    const float*     W  = (const float*)d_in[2];
    const float*     b  = (const float*)d_in[3];
    float*           out = (float*)d_out;

    const long long* row = ei;            // edge_index[0]
    const long long* col = ei + N_EDGES;  // edge_index[1]

    // workspace layout (16B-aligned sections):
    //   [0 .. N)        unsigned deg counters
    //   [N .. 2N)       float dinv
    //   [2N .. 2N+N*D)  float xw
    unsigned int* deg  = (unsigned int*)d_ws;
    float*        dinv = (float*)d_ws + N_NODES;
    float*        xw   = (float*)d_ws + 2 * N_NODES;

    const int total = N_NODES * DIM;                 // 6,400,000
    const int bsz   = 256;

    gcn_init<<<(total + bsz - 1) / bsz, bsz, 0, stream>>>(out, deg);
    gcn_degree<<<(N_EDGES + bsz - 1) / bsz, bsz, 0, stream>>>(col, deg);
    gcn_dinv<<<(N_NODES + bsz - 1) / bsz, bsz, 0, stream>>>(deg, dinv);
    gcn_gemm_wmma<<<N_NODES / 16, bsz, 0, stream>>>(x, W, xw);   // 3125 blocks
    {
        long long sthreads = (long long)N_EDGES * 32;            // 25.6M
        int sblocks = (int)((sthreads + bsz - 1) / bsz);         // 100,000
        gcn_scatter<<<sblocks, bsz, 0, stream>>>(row, col, dinv, xw, out);
    }
    gcn_finalize<<<(total + bsz - 1) / bsz, bsz, 0, stream>>>(xw, dinv, b, out);
}